// RotaryPositionalEmbedding_57518202028109
// MI455X (gfx1250) — compile-verified
//
#include <hip/hip_runtime.h>
#include <hip/hip_bf16.h>
#include <stdint.h>

// RoPE: out[..., 2k]   = c*x[2k] - s*x[2k+1]
//       out[..., 2k+1] = s*x[2k] + c*x[2k+1]
// with c/s gathered from cos/sin tables at token_positions[seq].
//
// Shapes: x (4,16,4096,128) f32; token_positions (4096) i32;
//         cos_table/sin_table (8192,64) f32. Output same shape as x.
//
// Roofline: ~100 MFLOP vs ~270 MB HBM traffic -> pure bandwidth kernel.
// Strategy: stage per-tile cos/sin rows into LDS with CDNA5 async-to-LDS DMA
// (gather by token position happens in the per-lane global address), then
// stream x with nontemporal b128 loads/stores.

typedef float v4f __attribute__((ext_vector_type(4)));
typedef float v2f __attribute__((ext_vector_type(2)));

namespace {
constexpr int kB = 4;
constexpr int kH = 16;
constexpr int kS = 4096;
constexpr int kD = 128;
constexpr int kHalf = kD / 2;          // 64 table entries per position
constexpr int kPosTile = 128;          // seq positions per block
constexpr int kThreads = 256;          // 8 wave32s
constexpr int kTilesPerSeq = kS / kPosTile;        // 32
constexpr int kF4PerTile = kPosTile * kD / 4;      // 4096 float4 per block
constexpr int kXfersPerRow = kHalf / 4;            // 16 x 16B per table row
}

// One 16-byte global->LDS async copy (GLOBAL_LOAD_ASYNC_TO_LDS_B128, ASYNCcnt).
// lds_byte = wave-relative LDS byte address (low 32 bits of flat shared addr).
__device__ __forceinline__ void async_copy_b128(uint32_t lds_byte, const void* gaddr) {
    asm volatile("global_load_async_to_lds_b128 %0, %1, off"
                 :
                 : "v"(lds_byte), "v"(gaddr)
                 : "memory");
}

__device__ __forceinline__ void wait_asynccnt0() {
    asm volatile("s_wait_asynccnt 0" ::: "memory");
}

__global__ __launch_bounds__(kThreads) void rope_fwd_kernel(
    const float* __restrict__ x,
    const int* __restrict__ token_positions,
    const float* __restrict__ cos_table,
    const float* __restrict__ sin_table,
    float* __restrict__ out) {
    // 32 KiB each: cos/sin for this block's 128 positions (64 KiB/block ->
    // up to 5 blocks per 320 KiB WGP).
    __shared__ float c_sm[kPosTile * kHalf];
    __shared__ float s_sm[kPosTile * kHalf];

    const int tile = blockIdx.x & (kTilesPerSeq - 1);
    const int bh   = blockIdx.x / kTilesPerSeq;   // 0..63 (batch*heads)
    const int pos0 = tile * kPosTile;
    const int t    = threadIdx.x;

    // ---- Stage cos/sin tile into LDS via async DMA ----
    // Thread t: position p = t & 127, table = t >> 7 (waves 0-3 cos, 4-7 sin).
    // Exactly ONE coalesced token_positions load per thread, then a burst of
    // 16 async b128 transfers covering the whole 64-float row (no dependent
    // load->wait chains; 16 outstanding async ops per wave, well under 63).
    {
        const int p     = t & (kPosTile - 1);
        const int tab   = t >> 7;                       // 0 = cos, 1 = sin
        const int tp    = token_positions[pos0 + p];    // single load
        const float* gsrc = (tab ? sin_table : cos_table) + (size_t)tp * kHalf;
        float* lbase      = (tab ? s_sm : c_sm) + p * kHalf;
        const uint32_t l0 = (uint32_t)(uintptr_t)lbase;
        #pragma unroll
        for (int c4 = 0; c4 < kXfersPerRow; ++c4) {
            async_copy_b128(l0 + (uint32_t)(c4 * 16), gsrc + c4 * 4);
        }
    }
    wait_asynccnt0();   // this wave's async LDS writes complete
    __syncthreads();    // visible to all waves in the block

    // ---- Stream x -> rotate -> out, 16B per lane per iteration ----
    const size_t rowBase = ((size_t)bh * kS + (size_t)pos0) * kD;
    #pragma unroll 4
    for (int it = 0; it < kF4PerTile / kThreads; ++it) {
        const int j = t + it * kThreads;   // 0..4095
        const int p = j >> 5;              // position within tile (32 f4/row)
        const int f = j & 31;              // f4 index within the 128-dim row
        const size_t off = rowBase + (size_t)p * kD + (size_t)(4 * f);

        // x read once -> nontemporal load
        v4f v = __builtin_nontemporal_load((const v4f*)(x + off));

        // pair indices 2f, 2f+1 -> contiguous float2 in LDS (conflict-free:
        // 32 lanes x 8B = one full 64-bank x 4B sweep)
        const int ti = p * kHalf + 2 * f;
        v2f c2 = *(const v2f*)&c_sm[ti];
        v2f s2 = *(const v2f*)&s_sm[ti];

        v4f o;
        o.x = c2.x * v.x - s2.x * v.y;
        o.y = s2.x * v.x + c2.x * v.y;
        o.z = c2.y * v.z - s2.y * v.w;
        o.w = s2.y * v.z + c2.y * v.w;

        // out written once, never re-read -> nontemporal store
        __builtin_nontemporal_store(o, (v4f*)(out + off));
    }
}

extern "C" void kernel_launch(void* const* d_in, const int* in_sizes, int n_in,
                              void* d_out, int out_size, void* d_ws, size_t ws_size,
                              hipStream_t stream) {
    const float* x   = (const float*)d_in[0];
    const int*   tp  = (const int*)d_in[1];
    const float* ct  = (const float*)d_in[2];
    const float* st  = (const float*)d_in[3];
    float*       out = (float*)d_out;

    dim3 grid(kB * kH * kTilesPerSeq);   // 2048 blocks
    dim3 block(kThreads);                // 256 threads = 8 wave32s
    rope_fwd_kernel<<<grid, block, 0, stream>>>(x, tp, ct, st, out);
}